// VectorQuantizer_27633819582652
// MI455X (gfx1250) — compile-verified
//
#include <hip/hip_runtime.h>
#include <hip/hip_bf16.h>

typedef __attribute__((ext_vector_type(2))) float v2f;
typedef __attribute__((ext_vector_type(8))) float v8f;
typedef __attribute__((ext_vector_type(4))) int   v4i;

#define DIMS      256
#define TLEN      1024
#define NBATCH    16
#define NROWS     (NBATCH * TLEN)   // 16384
#define KENT      8192
#define ROWS_PER_WG 128
#define NWAVES    8
#define ASTRIDE   260               // 256 + 4 pad: conflict-free fragment reads
#define STAGE_ENT 32                // codebook entries per double-buffered stage
#define NSTAGES   (KENT / STAGE_ENT) // 256

#if __has_builtin(__builtin_amdgcn_global_load_async_to_lds_b128) && \
    __has_builtin(__builtin_amdgcn_s_wait_asynccnt)
#define USE_ASYNC 1
#else
#define USE_ASYNC 0
#endif

#if USE_ASYNC
typedef v4i __attribute__((address_space(1)))* gv4p;   // global src (AS1)
typedef v4i __attribute__((address_space(3)))* lv4p;   // LDS dst (AS3)
#endif

// copy 16 consecutive floats global -> LDS (async if available)
__device__ __forceinline__ void cp16(const float* gp, float* lp) {
#if USE_ASYNC
    __builtin_amdgcn_global_load_async_to_lds_b128((gv4p)(gp + 0),  (lv4p)(lp + 0),  0, 0);
    __builtin_amdgcn_global_load_async_to_lds_b128((gv4p)(gp + 4),  (lv4p)(lp + 4),  0, 0);
    __builtin_amdgcn_global_load_async_to_lds_b128((gv4p)(gp + 8),  (lv4p)(lp + 8),  0, 0);
    __builtin_amdgcn_global_load_async_to_lds_b128((gv4p)(gp + 12), (lv4p)(lp + 12), 0, 0);
#else
    float4 v0 = ((const float4*)gp)[0];
    float4 v1 = ((const float4*)gp)[1];
    float4 v2 = ((const float4*)gp)[2];
    float4 v3 = ((const float4*)gp)[3];
    float4* d = (float4*)lp;
    d[0] = v0; d[1] = v1; d[2] = v2; d[3] = v3;
#endif
}

__device__ __forceinline__ void cp4(const float* gp, float* lp) {
#if USE_ASYNC
    __builtin_amdgcn_global_load_async_to_lds_b128((gv4p)gp, (lv4p)lp, 0, 0);
#else
    *(float4*)lp = *(const float4*)gp;
#endif
}

__device__ __forceinline__ void async_fence_all() {
#if USE_ASYNC
    __builtin_amdgcn_s_wait_asynccnt(0);
#endif
    __syncthreads();
}

// ---------------- Phase 1: codebook squared norms ----------------
__global__ void vq_enorm(const float* __restrict__ cb, float* __restrict__ enorm) {
    const int lane = threadIdx.x & 31;
    const int row  = blockIdx.x * 8 + (threadIdx.x >> 5);
    const float* rp = cb + (size_t)row * DIMS + lane * 8;
    float4 a = ((const float4*)rp)[0];
    float4 b = ((const float4*)rp)[1];
    float s = a.x*a.x + a.y*a.y + a.z*a.z + a.w*a.w
            + b.x*b.x + b.y*b.y + b.z*b.z + b.w*b.w;
    #pragma unroll
    for (int off = 16; off >= 1; off >>= 1) s += __shfl_xor(s, off, 32);
    if (lane == 0) enorm[row] = s;
}

// ---------------- Phase 2: fused GEMM + argmin + gather + loss ----------------
__launch_bounds__(256)
__global__ void vq_main(const float* __restrict__ z, const float* __restrict__ cb,
                        const float* __restrict__ enorm, float* __restrict__ outq,
                        float* __restrict__ partial) {
    __shared__ __align__(16) float a_lds[NWAVES][16][ASTRIDE];      // 133,120 B
    __shared__ __align__(16) float b_lds[2][STAGE_ENT][ASTRIDE];    //  66,560 B (double buffer)
    __shared__ __align__(16) float e_lds[KENT];                     //  32,768 B
    __shared__ int   idx_lds[ROWS_PER_WG];
    __shared__ float wg_sum[NWAVES];

    const int tid  = threadIdx.x;
    const int wave = tid >> 5;
    const int lane = tid & 31;
    const int n0   = blockIdx.x * ROWS_PER_WG;     // 128 consecutive rows, same batch b
    const int b    = n0 / TLEN;
    const int t0   = n0 % TLEN;

    // ---- stage A: transpose z[b][d][t0..t0+127] into per-wave row-major tiles
    {
        const int t_local = tid & 127;
        const int d_start = tid >> 7;              // 0..1
        const float* zp = z + (size_t)b * DIMS * TLEN + (size_t)t0 + t_local;
        for (int d = d_start; d < DIMS; d += 2)
            a_lds[t_local >> 4][t_local & 15][d] = zp[(size_t)d * TLEN];
    }

    // ---- preload codebook norms into LDS (async)
    {
        const float* ep = enorm + tid * 4;
        float* lp = e_lds + tid * 4;
        #pragma unroll
        for (int i = 0; i < 8; ++i)
            cp4(ep + i * 1024, lp + i * 1024);
    }

    // ---- B-stage cooperative load mapping: 16 threads/row, 16 floats each
    const int rr = tid >> 4;                       // 0..15 (rows rr and rr+16)
    const int c0 = (tid & 15) * 16;

    // ---- prologue: stage 0 into buffer 0
    {
        const float* cp = cb + (size_t)rr * DIMS + c0;
        cp16(cp,                       &b_lds[0][rr][c0]);
        cp16(cp + (size_t)16 * DIMS,   &b_lds[0][16 + rr][c0]);
    }
    async_fence_all();

    float minv[8];
    int   mini[8];
    #pragma unroll
    for (int r = 0; r < 8; ++r) { minv[r] = 3.4e38f; mini[r] = 0; }

    const int colA = lane & 15;          // C column / codebook entry within tile
    const int koff = (lane >> 4) * 2;    // K offset for A/B fragments

    for (int s = 0; s < NSTAGES; ++s) {
        // ---- prefetch next stage into the other buffer (async, overlapped with WMMAs)
        if (s + 1 < NSTAGES) {
            const int nbuf = (s + 1) & 1;
            const float* cp = cb + (size_t)((s + 1) * STAGE_ENT + rr) * DIMS + c0;
            cp16(cp,                     &b_lds[nbuf][rr][c0]);
            cp16(cp + (size_t)16 * DIMS, &b_lds[nbuf][16 + rr][c0]);
        }

        // ---- two independent 16x16 WMMA chains sharing the A fragment
        const int buf = s & 1;
        v8f acc0 = {}, acc1 = {};
        const float* ap = &a_lds[wave][colA][koff];
        const float* b0 = &b_lds[buf][colA][koff];
        const float* b1 = &b_lds[buf][16 + colA][koff];
        #pragma unroll
        for (int d0 = 0; d0 < DIMS; d0 += 4) {
            v2f av  = *(const v2f*)(ap + d0);
            v2f bv0 = *(const v2f*)(b0 + d0);
            v2f bv1 = *(const v2f*)(b1 + d0);
            acc0 = __builtin_amdgcn_wmma_f32_16x16x4_f32(
                false, av, false, bv0, (short)0, acc0, false, false);
            acc1 = __builtin_amdgcn_wmma_f32_16x16x4_f32(
                false, av, false, bv1, (short)0, acc1, false, false);
        }

        // ---- streaming argmin: dist = ||e||^2 - 2 z.e  (||z||^2 constant per row)
        const int   e0  = s * STAGE_ENT + colA;
        const float en0 = e_lds[e0];
        const float en1 = e_lds[e0 + 16];
        #pragma unroll
        for (int r = 0; r < 8; ++r) {
            float d0v = en0 - 2.0f * acc0[r];
            if (d0v < minv[r]) { minv[r] = d0v; mini[r] = e0; }
            float d1v = en1 - 2.0f * acc1[r];
            if (d1v < minv[r]) { minv[r] = d1v; mini[r] = e0 + 16; }
        }

        // ---- wait for prefetch, then swap buffers
        async_fence_all();
    }

    // ---- cross-lane argmin within each 16-lane half (cols of the C tile)
    #pragma unroll
    for (int r = 0; r < 8; ++r) {
        float v = minv[r];
        int   i = mini[r];
        #pragma unroll
        for (int off = 8; off >= 1; off >>= 1) {
            float ov = __shfl_xor(v, off, 32);
            int   oi = __shfl_xor(i, off, 32);
            if (ov < v || (ov == v && oi < i)) { v = ov; i = oi; }
        }
        if ((lane & 15) == 0) {
            const int row = r + (lane >> 4) * 8;   // lanes0-15 -> row r, lanes16-31 -> row r+8
            idx_lds[wave * 16 + row] = i;
        }
    }
    __syncthreads();

    // ---- gather quantized output (back to [B, D, T]) + loss partial
    float lsum = 0.0f;
    {
        const int t_local = tid & 127;
        const int d_start = tid >> 7;
        const int qrow = idx_lds[t_local];
        const float* qp = cb + (size_t)qrow * DIMS;
        const float* al = &a_lds[t_local >> 4][t_local & 15][0];
        float* op = outq + (size_t)b * DIMS * TLEN + (size_t)t0 + t_local;
        for (int d = d_start; d < DIMS; d += 2) {
            float q  = qp[d];
            float zz = al[d];
            float df = q - zz;
            lsum += df * df;
            op[(size_t)d * TLEN] = q;
        }
    }
    #pragma unroll
    for (int off = 16; off >= 1; off >>= 1) lsum += __shfl_xor(lsum, off, 32);
    if (lane == 0) wg_sum[wave] = lsum;
    __syncthreads();
    if (tid == 0) {
        float s = 0.0f;
        #pragma unroll
        for (int w = 0; w < NWAVES; ++w) s += wg_sum[w];
        partial[blockIdx.x] = s;
    }
}

// ---------------- Phase 3: deterministic loss finalize ----------------
__global__ void vq_finalize(const float* __restrict__ partial, float* __restrict__ loss_out) {
    float s = 0.0f;
    for (int i = 0; i < NROWS / ROWS_PER_WG; ++i) s += partial[i];
    // loss = q_latent + 0.25 * e_latent = 1.25 * mean((q - z)^2)
    loss_out[0] = 1.25f * s / (float)((size_t)NROWS * DIMS);
}

extern "C" void kernel_launch(void* const* d_in, const int* in_sizes, int n_in,
                              void* d_out, int out_size, void* d_ws, size_t ws_size,
                              hipStream_t stream) {
    (void)in_sizes; (void)n_in; (void)out_size; (void)ws_size;
    const float* z  = (const float*)d_in[0];   // [16, 256, 1024]
    const float* cb = (const float*)d_in[1];   // [8192, 256]
    float* out    = (float*)d_out;             // [16,256,1024] quantized + 1 loss
    float* enorm  = (float*)d_ws;              // 8192 floats
    float* part   = enorm + KENT;              // 128 floats

    vq_enorm<<<KENT / 8, 256, 0, stream>>>(cb, enorm);
    vq_main<<<NROWS / ROWS_PER_WG, 256, 0, stream>>>(z, cb, enorm, out, part);
    vq_finalize<<<1, 1, 0, stream>>>(part, out + (size_t)NROWS * DIMS);
}